// GCN_82540681494813
// MI455X (gfx1250) — compile-verified
//
#include <hip/hip_runtime.h>

typedef __attribute__((ext_vector_type(2))) float v2f;
typedef __attribute__((ext_vector_type(8))) float v8f;

// ---------------------------------------------------------------------------
// Small elementwise helpers
// ---------------------------------------------------------------------------
__global__ void k_fill(float* __restrict__ p, float val, int n) {
  int i = blockIdx.x * blockDim.x + threadIdx.x;
  if (i < n) p[i] = val;
}

__global__ void k_deg_accum(const int* __restrict__ dst, float* __restrict__ deg, int E) {
  int i = blockIdx.x * blockDim.x + threadIdx.x;
  if (i < E) unsafeAtomicAdd(&deg[dst[i]], 1.0f);   // counts < 2^24 -> exact in f32
}

__global__ void k_rsqrt_inplace(float* __restrict__ d, int n) {
  int i = blockIdx.x * blockDim.x + threadIdx.x;
  if (i < n) d[i] = 1.0f / sqrtf(d[i]);             // precise path (no v_rsq approx)
}

__global__ void k_relu(float* __restrict__ p, int n) {
  int i = blockIdx.x * blockDim.x + threadIdx.x;
  if (i < n) p[i] = fmaxf(p[i], 0.0f);
}

// agg[i,f] = h[i,f] * dinv[i]^2 + bias[f]   (self-loop term + bias, init for scatter)
__global__ void k_init_agg(const float* __restrict__ h, const float* __restrict__ dinv,
                           const float* __restrict__ bias, float* __restrict__ agg,
                           int N, int F) {
  int i = blockIdx.x * blockDim.x + threadIdx.x;
  if (i >= N * F) return;
  int node = i / F;
  int f    = i - node * F;
  float di = dinv[node];
  agg[i] = h[i] * (di * di) + bias[f];
}

// ---------------------------------------------------------------------------
// Dense GEMM via V_WMMA_F32_16X16X4_F32 (exact fp32, wave32, 16x16 tile/wave)
// A: [M x K] row-major, B: [K x Nc] row-major, C: [M x Nc].
// Lane layout per CDNA5 ISA (32-bit A 16x4, 32-bit C/D 16x16):
//   lane = half*16 + j
//   A vgpr v : A[j][k + 2*half + v]      (contiguous -> float2 load)
//   B vgpr v : B[k + 2*half + v][j]      (stride-Nc loads)
//   D vgpr v : D[v + 8*half][j]
// ---------------------------------------------------------------------------
__global__ __launch_bounds__(256)
void k_gemm_wmma(const float* __restrict__ A, const float* __restrict__ B,
                 float* __restrict__ C, int M, int K, int Nc) {
  int wid  = (blockIdx.x * blockDim.x + threadIdx.x) >> 5;
  int lane = threadIdx.x & 31;
  int tilesN = Nc >> 4;
  int tilesM = M >> 4;
  if (wid >= tilesM * tilesN) return;              // wave-uniform: EXEC stays all-1s
  int tm = wid / tilesN;
  int tn = wid - tm * tilesN;

  int half = lane >> 4;                            // 0 or 1
  int j    = lane & 15;

  const float* Arow = A + (size_t)(tm * 16 + j) * K + 2 * half;   // row j, K-offset 2*half
  const float* Bp   = B + (size_t)(2 * half) * Nc + tn * 16 + j;  // col j, K-offset 2*half

  v8f acc = {};
  for (int k = 0; k < K; k += 4) {
    v2f a = *(const v2f*)(Arow + k);
    v2f b;
    b.x = Bp[(size_t)k * Nc];
    b.y = Bp[(size_t)(k + 1) * Nc];
    // 8 args: (neg_a, A, neg_b, B, c_mod, C, reuse_a, reuse_b)
    acc = __builtin_amdgcn_wmma_f32_16x16x4_f32(false, a, false, b, (short)0, acc,
                                                false, false);
  }

  float* Cp = C + (size_t)(tm * 16 + 8 * half) * Nc + tn * 16 + j;
#pragma unroll
  for (int v = 0; v < 8; ++v) Cp[(size_t)v * Nc] = acc[v];
}

// ---------------------------------------------------------------------------
// Edge scatter: one wave per edge. Lane-strided coalesced gather of h[src]
// (hits the 192MB L2 — table is only 25.6MB), scaled by dinv[s]*dinv[d],
// hardware f32 atomic add into agg[dst].
// ---------------------------------------------------------------------------
__global__ __launch_bounds__(256)
void k_scatter(const float* __restrict__ h, const int* __restrict__ src,
               const int* __restrict__ dst, const float* __restrict__ dinv,
               float* __restrict__ agg, int E, int F) {
  int gid  = blockIdx.x * blockDim.x + threadIdx.x;
  int e    = gid >> 5;
  int lane = gid & 31;
  if (e >= E) return;
  int s = src[e], d = dst[e];
  float norm = dinv[s] * dinv[d];
  const float* hs = h + (size_t)s * F;
  float* ad       = agg + (size_t)d * F;
  for (int f = lane; f < F; f += 32)
    unsafeAtomicAdd(&ad[f], hs[f] * norm);
}

// ---------------------------------------------------------------------------
extern "C" void kernel_launch(void* const* d_in, const int* in_sizes, int n_in,
                              void* d_out, int out_size, void* d_ws, size_t ws_size,
                              hipStream_t stream) {
  const float* x  = (const float*)d_in[0];
  const int*   ei = (const int*)d_in[1];
  const float* W1 = (const float*)d_in[2];
  const float* b1 = (const float*)d_in[3];
  const float* W2 = (const float*)d_in[4];
  const float* b2 = (const float*)d_in[5];
  float* out = (float*)d_out;

  const int F0 = 128;
  const int N  = in_sizes[0] / F0;   // 50000
  const int E  = in_sizes[1] / 2;    // 1600000
  const int F1 = in_sizes[3];        // 128
  const int F2 = in_sizes[5];        // 64
  const int* src = ei;
  const int* dst = ei + E;

  // Workspace: dinv[N] | bufA[N*F1] (h1, later h2) | bufB[N*F1] (agg1 -> hact)
  float* dinv = (float*)d_ws;
  float* h1   = dinv + N;
  float* agg1 = h1 + (size_t)N * F1;
  float* h2   = h1;                   // reuse: h1 dead once layer-1 scatter finishes

  const int B = 256;
  auto cdiv = [](long long a, long long b) { return (int)((a + b - 1) / b); };

  // --- degree / normalization -------------------------------------------------
  k_fill<<<cdiv(N, B), B, 0, stream>>>(dinv, 1.0f, N);               // self-loop
  k_deg_accum<<<cdiv(E, B), B, 0, stream>>>(dst, dinv, E);
  k_rsqrt_inplace<<<cdiv(N, B), B, 0, stream>>>(dinv, N);

  // --- layer 1: h1 = x@W1 ; agg1 = scatter + self-loop + bias ; relu ---------
  long long waves1 = (long long)(N / 16) * (F1 / 16);
  k_gemm_wmma<<<cdiv(waves1 * 32, B), B, 0, stream>>>(x, W1, h1, N, F0, F1);
  k_init_agg<<<cdiv((long long)N * F1, B), B, 0, stream>>>(h1, dinv, b1, agg1, N, F1);
  k_scatter<<<cdiv((long long)E * 32, B), B, 0, stream>>>(h1, src, dst, dinv, agg1, E, F1);
  k_relu<<<cdiv((long long)N * F1, B), B, 0, stream>>>(agg1, N * F1);

  // --- layer 2: h2 = relu(agg1)@W2 ; aggregate straight into d_out -----------
  long long waves2 = (long long)(N / 16) * (F2 / 16);
  k_gemm_wmma<<<cdiv(waves2 * 32, B), B, 0, stream>>>(agg1, W2, h2, N, F1, F2);
  k_init_agg<<<cdiv((long long)N * F2, B), B, 0, stream>>>(h2, dinv, b2, out, N, F2);
  k_scatter<<<cdiv((long long)E * 32, B), B, 0, stream>>>(h2, src, dst, dinv, out, E, F2);
}